// PositionSensitiveScoreMap_44710609551741
// MI455X (gfx1250) — compile-verified
//
#include <hip/hip_runtime.h>
#include <hip/hip_bf16.h>

typedef __attribute__((ext_vector_type(2))) float v2f;
typedef __attribute__((ext_vector_type(4))) float v4f;
typedef __attribute__((ext_vector_type(8))) float v8f;

#define KP 7
#define C1 21
#define KK 49        // KP*KP
#define HF 128
#define WF 128
#define NCH 1029     // KK*C1
#define FEAT_STRIDE 30
#define NROI 2000

// D = A(16x4 f32) * B(4x16 f32) + C ; chained x4 gives 16x16x16
__device__ __forceinline__ v8f wmma4(v2f a, v2f b, v8f c) {
  return __builtin_amdgcn_wmma_f32_16x16x4_f32(false, a, false, b, (short)0, c,
                                               false, false);
}

// One workgroup = one channel (kk,c). Builds exclusive 2-D prefix sum of the
// 128x128 map via triangular WMMA matmuls, stores channel-interleaved:
//   St[((kk*128 + y)*128 + x)*21 + c] = sum_{yy<y, xx<x} A[yy][xx]
__global__ void __launch_bounds__(256) sat_kernel(const float* __restrict__ in,
                                                  float* __restrict__ St) {
  __shared__ float tile[HF * WF];  // 64 KB
  const int ch = blockIdx.x;       // ch = kk*21 + c  (matches reshape order)
  const int kk = ch / C1;
  const int c  = ch - kk * C1;
  const int t  = threadIdx.x;

  // Stream the channel into LDS with NT loads (read-once; keep L2 for St).
  const v4f* src = (const v4f*)(in + (size_t)ch * (HF * WF));
  v4f* dst4 = (v4f*)tile;
#pragma unroll
  for (int i = 0; i < 16; ++i) {
    const int idx = t + i * 256;
    dst4[idx] = __builtin_nontemporal_load(&src[idx]);
  }
  __syncthreads();

  const int wv   = t >> 5;         // 8 waves
  const int lane = t & 31;
  const int Lm   = lane & 15;
  const int hi   = lane >> 4;      // 0 or 1

  // Strictly-triangular ones constant: element (K = 4s + 2*hi + v) vs Lm.
  // Same lane expression serves as B (upper-tri, K<N) in phase 1 and as
  // A (lower-tri, K<M) in phase 2 given the respective layouts.
  v2f tri[4], ones = {1.f, 1.f};
#pragma unroll
  for (int s = 0; s < 4; ++s) {
    const int k0 = 4 * s + 2 * hi;
    tri[s][0] = (k0 + 0 < Lm) ? 1.f : 0.f;
    tri[s][1] = (k0 + 1 < Lm) ? 1.f : 0.f;
  }

  // -------- Phase 1: row exclusive prefix (in place). Wave owns 16 rows.
  {
    v8f carry = {};                          // per-row carry, replicated in N
    const int rowA = (wv << 4) + Lm;         // A-layout: M on lanes
    for (int jb = 0; jb < 8; ++jb) {
      v2f a[4];
#pragma unroll
      for (int s = 0; s < 4; ++s) {          // A-layout: K = 4s + 2*hi + v
        const int col = (jb << 4) + 4 * s + 2 * hi;
        a[s][0] = tile[rowA * WF + col];
        a[s][1] = tile[rowA * WF + col + 1];
      }
      v8f d = carry;                         // carry folded into C operand
#pragma unroll
      for (int s = 0; s < 4; ++s) d = wmma4(a[s], tri[s], d);      // T * U_s
#pragma unroll
      for (int s = 0; s < 4; ++s) carry = wmma4(a[s], ones, carry); // += rowsum
#pragma unroll
      for (int r = 0; r < 8; ++r) {          // D-layout store
        const int y = (wv << 4) + r + 8 * hi;
        const int x = (jb << 4) + Lm;
        tile[y * WF + x] = d[r];
      }
    }
  }
  __syncthreads();

  // -------- Phase 2: column exclusive prefix; wave owns 16 columns.
  {
    v8f carry = {};                          // per-col carry, replicated in M
    const int jb = wv;
    float* outbase = St + (size_t)kk * (HF * WF * C1) + c;
    for (int ib = 0; ib < 8; ++ib) {
      v2f b[4];
#pragma unroll
      for (int s = 0; s < 4; ++s) {          // B-layout: K = 4s + 2*hi + v
        const int row = (ib << 4) + 4 * s + 2 * hi;
        const int x   = (jb << 4) + Lm;
        b[s][0] = tile[(row + 0) * WF + x];
        b[s][1] = tile[(row + 1) * WF + x];
      }
      v8f e = carry;
#pragma unroll
      for (int s = 0; s < 4; ++s) e = wmma4(tri[s], b[s], e);       // L_s * X
#pragma unroll
      for (int s = 0; s < 4; ++s) carry = wmma4(ones, b[s], carry); // += colsum
#pragma unroll
      for (int r = 0; r < 8; ++r) {
        const int y = (ib << 4) + r + 8 * hi;
        const int x = (jb << 4) + Lm;
        outbase[(size_t)(y * WF + x) * C1] = e[r];                  // L2-resident
      }
    }
  }
}

// One workgroup per ROI. Deterministic 12-group x 21-class partial sums,
// fixed-order reduction, softmax over 21 classes.
__global__ void __launch_bounds__(256) pool_kernel(const float* __restrict__ St,
                                                   const int* __restrict__ rois,
                                                   float* __restrict__ out) {
  __shared__ float part[12 * C1];
  __shared__ float red[C1];
  const int n = blockIdx.x;
  const int t = threadIdx.x;

  const int y0 = rois[4 * n + 0] / FEAT_STRIDE;
  const int x0 = rois[4 * n + 1] / FEAT_STRIDE;
  const int y1 = rois[4 * n + 2] / FEAT_STRIDE;
  const int x1 = rois[4 * n + 3] / FEAT_STRIDE;
  const int ystep = (y1 - y0) / KP;
  const int xstep = (x1 - x0) / KP;

  if (t < 12 * C1) {
    const int c = t % C1;       // adjacent lanes -> adjacent c -> 84B runs
    const int g = t / C1;
    float partial = 0.f;
    for (int jl = g; jl < KK; jl += 12) {
      const int j  = jl / KP;
      const int l  = jl - KP * j;
      const int ys = y0 + j * ystep, ye = ys + ystep;
      const int xs = x0 + l * xstep, xe = xs + xstep;
      const float* base = St + (size_t)jl * (HF * WF) * C1 + c;
      const float see = base[(size_t)(ye * WF + xe) * C1];
      const float sse = base[(size_t)(ys * WF + xe) * C1];
      const float ses = base[(size_t)(ye * WF + xs) * C1];
      const float sss = base[(size_t)(ys * WF + xs) * C1];
      partial += (see - sse) - (ses - sss);
    }
    part[t] = partial;
  }
  __syncthreads();
  if (t < C1) {
    float s = 0.f;
#pragma unroll
    for (int g = 0; g < 12; ++g) s += part[g * C1 + t];
    const float area = (float)(ystep * xstep);
    const float inv  = (area > 0.f) ? 1.f / (area * (float)KK) : 0.f;
    red[t] = s * inv;
  }
  __syncthreads();
  if (t == 0) {
    float m = red[0];
    for (int cc = 1; cc < C1; ++cc) m = fmaxf(m, red[cc]);
    float e[C1];
    float sum = 0.f;
    for (int cc = 0; cc < C1; ++cc) { e[cc] = __expf(red[cc] - m); sum += e[cc]; }
    const float r = 1.f / sum;
    for (int cc = 0; cc < C1; ++cc) out[(size_t)n * C1 + cc] = e[cc] * r;
  }
}

extern "C" void kernel_launch(void* const* d_in, const int* in_sizes, int n_in,
                              void* d_out, int out_size, void* d_ws, size_t ws_size,
                              hipStream_t stream) {
  const float* cls = (const float*)d_in[0];   // (1, 1029, 128, 128) f32
  const int*   roi = (const int*)d_in[1];     // (2000, 4) i32
  float*       out = (float*)d_out;           // (2000, 21) f32
  float*       St  = (float*)d_ws;            // needs 1029*16384*4 = 67.4 MB

  sat_kernel<<<NCH, 256, 0, stream>>>(cls, St);
  pool_kernel<<<NROI, 256, 0, stream>>>(St, roi, out);
}